// OutputHead_11244224381106
// MI455X (gfx1250) — compile-verified
//
#include <hip/hip_runtime.h>
#include <hip/hip_bf16.h>

// out[b,h,i,j,d] = 0.5*(a+b) + a*b + e
//   a = seq[b,h,j,d], b = seq[b,h,i,d], e = emb[|i-j|,d]
// Refactored: out = fma(a, b+0.5, e + 0.5*b)   (b+0.5, 0.5*b hoisted)
//   -> 2 VALU per element in the hot loop (1 add + 1 fma).
//
// fp32 shapes: seq (2,4,256,256)=2MB, emb (256,256)=256KB,
// out (2,4,256,256,256)=536.9MB. Store-bandwidth bound: ~23us @ 23.3 TB/s.
// One workgroup per (bh,i): stage x2 row via gfx1250 async DMA into LDS,
// L2-resident 128-bit loads, contiguous non-temporal 128-bit stores.

typedef float v4f __attribute__((ext_vector_type(4)));

#define BINS    256
#define EMB_DIM 256

__global__ __launch_bounds__(256) void outputhead_kernel(
    const float* __restrict__ seq,   // (8, 256, 256)      [bh, row, d]
    const float* __restrict__ emb,   // (256, 256)         [dist, d]
    float* __restrict__ out)         // (8, 256, 256, 256) [bh, i, j, d]
{
    __shared__ float x2row[EMB_DIM];           // seq[bh][i][:] (1 KB)

    const int tid = threadIdx.x;               // 0..255
    const int wg  = blockIdx.x;                // 0..2047
    const int bh  = wg >> 8;                   // 0..7
    const int i   = wg & 255;                  // 0..255

    const float* seqbh   = seq + (size_t)bh * BINS * EMB_DIM;
    float*       outbase = out + ((size_t)(bh * BINS + i)) * BINS * EMB_DIM;

    // ---- Stage x2 = seq[bh][i][:] into LDS with the CDNA5 async-DMA path.
    // Threads 0..63 (waves 0,1; EXEC all-ones within each) move 16 B each:
    // GLOBAL_LOAD_ASYNC_TO_LDS_B128 writes LDS directly, tracked by ASYNCcnt.
    {
        const uint64_t gbase =
            (uint64_t)(uintptr_t)(seqbh + (size_t)i * EMB_DIM);
        if (tid < 64) {
            // Low 32 bits of the flat shared address == LDS byte offset.
            uint32_t lds_off =
                (uint32_t)(uintptr_t)(&x2row[0]) + (uint32_t)(tid * 16);
            uint64_t gaddr = gbase + (uint32_t)(tid * 16);
            asm volatile("global_load_async_to_lds_b128 %0, %1, off"
                         :: "v"(lds_off), "v"(gaddr)
                         : "memory");
        }
        asm volatile("s_wait_asynccnt 0" ::: "memory");   // waves w/o issue: cnt==0
        __syncthreads();
    }

    // Thread tiling: 4 j-rows x 64 float4 d-chunks per iteration step.
    const int jl = tid >> 6;                   // 0..3  (j sub-row)
    const int d  = (tid & 63) * 4;             // 0..252 (float4 along EMB_DIM)

    const v4f b4  = *(const v4f*)(&x2row[d]);  // LDS read (16B aligned)
    const v4f b4p = b4 + 0.5f;                 // hoisted (b + 0.5)
    const v4f h4  = b4 * 0.5f;                 // hoisted (0.5 * b)

    // Incremental pointers: both advance 4 rows (4*EMB_DIM floats) per step.
    const float* ap = seqbh   + (size_t)jl * EMB_DIM + d;
    float*       op = outbase + (size_t)jl * EMB_DIM + d;
    const float* ep = emb + d;

    int j = jl;
    #pragma unroll 8
    for (int it = 0; it < BINS / 4; ++it) {
        const int dj = (i > j) ? (i - j) : (j - i);        // |i-j| <= 255

        const v4f a4 = *(const v4f*)ap;
        const v4f e4 = *(const v4f*)(ep + ((size_t)dj << 8));  // dj*EMB_DIM

        // out = a*(b+0.5) + (e + 0.5*b)   == 0.5*(a+b) + a*b + e
        const v4f r = __builtin_elementwise_fma(a4, b4p, e4 + h4);

        // 537 MB output stream: NT so it never evicts the 2.3 MB of hot
        // inputs from L2; b128 fully coalesced (4 KB contiguous per block-step).
        __builtin_nontemporal_store(r, (v4f*)op);

        j  += 4;
        ap += 4 * EMB_DIM;
        op += 4 * EMB_DIM;
    }
}

extern "C" void kernel_launch(void* const* d_in, const int* in_sizes, int n_in,
                              void* d_out, int out_size, void* d_ws, size_t ws_size,
                              hipStream_t stream) {
    const float* seq = (const float*)d_in[0];   // (2,4,256,256) fp32
    const float* emb = (const float*)d_in[1];   // (256,256)     fp32
    float*       out = (float*)d_out;           // (2,4,256,256,256) fp32

    (void)in_sizes; (void)n_in; (void)out_size; (void)d_ws; (void)ws_size;

    dim3 grid(8 * BINS);   // one workgroup per (bh, i)
    dim3 block(256);       // 8 wave32 waves
    outputhead_kernel<<<grid, block, 0, stream>>>(seq, emb, out);
}